// GMNetwork_67181878444567
// MI455X (gfx1250) — compile-verified
//
#include <hip/hip_runtime.h>
#include <math.h>

// ---------------------------------------------------------------------------
// GMNetwork forward for MI455X (gfx1250, wave32).
// All GEMMs on v_wmma_f32_16x16x4_f32: full fp32 fidelity on the matrix pipe.
//
// B=16 T=256 R=116 D=128 H=8 DK=16 HID=512
// d_out = [ out (B*T*D=524288) | gm (B*T*T=1048576) | idxf (B*T=4096) ]
//
// f32 16x16x4 fragment layout (lane l, h=l/16, nl=l%16):
//   A (16x4):  a = { A[nl][k0+2h], A[nl][k0+2h+1] }          (one b64 load)
//   B (4x16):  b = { B[k0+2h][nl], B[k0+2h+1][nl] }          (pair-interleaved)
//   C/D (v8f): element v holds row M=v+8h, col N=nl
// ---------------------------------------------------------------------------

typedef __attribute__((ext_vector_type(2))) float v2f;
typedef __attribute__((ext_vector_type(8))) float v8f;

#define DEV static __device__ __forceinline__

DEV v8f wmma4(v2f a, v2f b, v8f c) {
  return __builtin_amdgcn_wmma_f32_16x16x4_f32(false, a, false, b, (short)0, c,
                                               false, false);
}

// ---------------------------------------------------------------------------
// Kernel 1: fused GMPool front-end (dominant ~31 GFLOP, dist never hits HBM).
// gm[b,i,j] = sigmoid( relu( bn( |X[b,i]-X[b,j]| @ W1^T + b1 ) ) @ W2^T + b2 )
// One block per (b,i); 8 waves x 2 j-tiles. K = 116 = 29 x 4 exactly.
// ---------------------------------------------------------------------------
__global__ void k_gmpool(const float* __restrict__ X, const float* __restrict__ W1,
                         const float* __restrict__ b1, const float* __restrict__ g_bn,
                         const float* __restrict__ be_bn, const float* __restrict__ W2,
                         const float* __restrict__ b2, float* __restrict__ gm) {
  extern __shared__ char smem[];
  float* sXi = (float*)smem;          // 120 (row i, padded)
  float* sX  = sXi + 120;             // 256 x 118 (row-pad vs LDS banks)
  float* sW  = sX + 256 * 118;        // W1 pair-interleaved: [58][128][2]
  const int b = blockIdx.y, i = blockIdx.x, tid = threadIdx.x;

  for (int u = tid; u < 120; u += 256)
    sXi[u] = (u < 116) ? X[(((size_t)b * 256) + i) * 116 + u] : 0.f;
  for (int u = tid; u < 256 * 116; u += 256)
    sX[(u / 116) * 118 + (u % 116)] = X[(size_t)b * 29696 + u];
  for (int u = tid; u < 128 * 116; u += 256) {
    int n = u / 116, k = u % 116;
    sW[(k >> 1) * 256 + (n << 1) + (k & 1)] = W1[u];
  }
  __syncthreads();

  const int wave = tid >> 5, l = tid & 31, h = l >> 4, nl = l & 15;
  const float bninv = rsqrtf(1.0f + 1e-5f);

  for (int jt = wave; jt < 16; jt += 8) {
    const int j0 = jt * 16;
    v8f acc[8] = {};
    for (int s = 0; s < 29; ++s) {
      const int k0 = 4 * s + 2 * h;
      v2f xj = *(const v2f*)&sX[(j0 + nl) * 118 + k0];
      v2f xi = *(const v2f*)&sXi[k0];
      float d0 = xj.x - xi.x, d1 = xj.y - xi.y;
      v2f a;
      a.x = sqrtf(d0 * d0 + 1e-9f);
      a.y = sqrtf(d1 * d1 + 1e-9f);
      const float* wrow = &sW[(2 * s + h) * 256];
#pragma unroll
      for (int nt = 0; nt < 8; ++nt) {
        v2f bb = *(const v2f*)&wrow[(nt * 16 + nl) << 1];
        acc[nt] = wmma4(a, bb, acc[nt]);
      }
    }
    // epilogue: BN -> ReLU -> dot W2 -> sigmoid ; rows live in 16-lane halves
    float part[8];
#pragma unroll
    for (int v = 0; v < 8; ++v) {
      float p = 0.f;
#pragma unroll
      for (int nt = 0; nt < 8; ++nt) {
        int n = nt * 16 + nl;
        float hh = acc[nt][v] + b1[n];
        hh = hh * (g_bn[n] * bninv) + be_bn[n];
        hh = fmaxf(hh, 0.f);
        p += hh * W2[n];
      }
#pragma unroll
      for (int m = 1; m < 16; m <<= 1) p += __shfl_xor(p, m, 32);
      part[v] = p;
    }
    if (nl == 0) {
#pragma unroll
      for (int v = 0; v < 8; ++v) {
        int j = j0 + v + 8 * h;
        float sg = 1.f / (1.f + expf(-(part[v] + b2[0])));
        if (j == i) sg = 1.0f;
        gm[(((size_t)b * 256) + i) * 256 + j] = sg;
      }
    }
  }
}

// ---------------------------------------------------------------------------
// Kernel 2: batched symmetric eigendecomposition (parallel two-sided Jacobi).
// Matrix resident in 256KB LDS (CDNA5 320KB WGP LDS); eigenvectors in ws.
// ---------------------------------------------------------------------------
__global__ void k_eigh(const float* __restrict__ gm, float* __restrict__ V,
                       float* __restrict__ Vs, float* __restrict__ vals) {
  extern __shared__ char smem[];
  float* sA   = (float*)smem;            // 256*256
  int*   sP   = (int*)(sA + 65536);      // 128
  int*   sQ   = sP + 128;                // 128
  float* sC   = (float*)(sQ + 128);      // 128
  float* sS   = sC + 128;                // 128
  float* sVal = sS + 128;                // 256
  int*   sOrd = (int*)(sVal + 256);      // 256
  const int b = blockIdx.x, tid = threadIdx.x;
  const float* G = gm + (size_t)b * 65536;
  float* Vb  = V  + (size_t)b * 65536;
  float* Vsb = Vs + (size_t)b * 65536;

  for (int u = tid; u < 65536; u += 256) sA[u] = G[u];
  for (int u = tid; u < 65536; u += 256) Vb[u] = ((u >> 8) == (u & 255)) ? 1.f : 0.f;
  __syncthreads();

  for (int sweep = 0; sweep < 10; ++sweep) {
    for (int r = 0; r < 255; ++r) {
      if (tid < 128) {
        int p, q;
        if (tid == 0) { p = 255; q = r % 255; }
        else { p = (r + tid) % 255; q = (r + 255 - tid) % 255; }
        if (p > q) { int t = p; p = q; q = t; }
        float app = sA[p * 256 + p], aqq = sA[q * 256 + q], apq = sA[p * 256 + q];
        float c = 1.f, s = 0.f;
        if (fabsf(apq) > 1e-12f) {
          float th = (aqq - app) / (2.f * apq);
          float t  = copysignf(1.f, th) / (fabsf(th) + sqrtf(th * th + 1.f));
          c = 1.f / sqrtf(t * t + 1.f);
          s = t * c;
        }
        sP[tid] = p; sQ[tid] = q; sC[tid] = c; sS[tid] = s;
      }
      __syncthreads();
      for (int u = tid; u < 32768; u += 256) {        // A <- A*J (columns)
        int m = u >> 8, row = u & 255;
        int p = sP[m], q = sQ[m]; float c = sC[m], s = sS[m];
        float x = sA[row * 256 + p], y = sA[row * 256 + q];
        sA[row * 256 + p] = c * x - s * y;
        sA[row * 256 + q] = s * x + c * y;
      }
      __syncthreads();
      for (int u = tid; u < 32768; u += 256) {        // A <- J^T*A (rows)
        int m = u >> 8, col = u & 255;
        int p = sP[m], q = sQ[m]; float c = sC[m], s = sS[m];
        float x = sA[p * 256 + col], y = sA[q * 256 + col];
        sA[p * 256 + col] = c * x - s * y;
        sA[q * 256 + col] = s * x + c * y;
      }
      __syncthreads();
      for (int u = tid; u < 32768; u += 256) {        // V <- V*J
        int m = u >> 8, row = u & 255;
        int p = sP[m], q = sQ[m]; float c = sC[m], s = sS[m];
        float x = Vb[(size_t)row * 256 + p], y = Vb[(size_t)row * 256 + q];
        Vb[(size_t)row * 256 + p] = c * x - s * y;
        Vb[(size_t)row * 256 + q] = s * x + c * y;
      }
      __syncthreads();
    }
  }
  sVal[tid] = sA[tid * 257];
  __syncthreads();
  {
    float lam = sVal[tid];
    int rk = 0;
    for (int j = 0; j < 256; ++j) {
      float vj = sVal[j];
      rk += (vj > lam) || (vj == lam && j < tid);   // stable descending rank
    }
    sOrd[rk] = tid;
  }
  __syncthreads();
  vals[b * 256 + tid] = sVal[sOrd[tid]];
  for (int u = tid; u < 65536; u += 256) {
    int row = u >> 8, rc = u & 255;
    Vsb[u] = Vb[(size_t)row * 256 + sOrd[rc]];
  }
}

// ---------------------------------------------------------------------------
// JAX threefry2x32 (key(42)) gumbel noise, bit-faithful counter layout.
// ---------------------------------------------------------------------------
DEV float gumbel_at(unsigned flat) {
  const unsigned nHalf = 524288u;  // (B*T*T)/2
  bool lo = flat < nHalf;
  unsigned c0 = lo ? flat : flat - nHalf;
  unsigned c1 = lo ? flat + nHalf : flat;
  const unsigned k0 = 0u, k1 = 42u;
  const unsigned ks2 = k0 ^ k1 ^ 0x1BD11BDAu;
  unsigned x0 = c0 + k0, x1 = c1 + k1;
#define TF_ROUND(r) { x0 += x1; x1 = (x1 << (r)) | (x1 >> (32 - (r))); x1 ^= x0; }
  TF_ROUND(13) TF_ROUND(15) TF_ROUND(26) TF_ROUND(6)
  x0 += k1;  x1 += ks2 + 1u;
  TF_ROUND(17) TF_ROUND(29) TF_ROUND(16) TF_ROUND(24)
  x0 += ks2; x1 += k0 + 2u;
  TF_ROUND(13) TF_ROUND(15) TF_ROUND(26) TF_ROUND(6)
  x0 += k0;  x1 += k1 + 3u;
  TF_ROUND(17) TF_ROUND(29) TF_ROUND(16) TF_ROUND(24)
  x0 += k1;  x1 += ks2 + 4u;
  TF_ROUND(13) TF_ROUND(15) TF_ROUND(26) TF_ROUND(6)
  x0 += ks2; x1 += k0 + 5u;
#undef TF_ROUND
  unsigned bits = lo ? x0 : x1;
  float f = __uint_as_float((bits >> 9) | 0x3F800000u) - 1.0f;  // [0,1)
  float u = f * (1.0f - 1e-20f) + 1e-20f;
  return -logf(-logf(u));
}

// ---------------------------------------------------------------------------
// Kernel 3: gumbel-softmax grouping -> idxf, me, ew. One wave per 32 rows.
// ---------------------------------------------------------------------------
__global__ void k_group(const float* __restrict__ Vs, const float* __restrict__ vals,
                        float* __restrict__ me, float* __restrict__ ew,
                        float* __restrict__ idxOut) {
  const int b = blockIdx.x;
  const int wave = threadIdx.x >> 5, l = threadIdx.x & 31;
  for (int t = wave * 32; t < wave * 32 + 32; ++t) {
    float vbuf[8];
    float m = -INFINITY;
#pragma unroll
    for (int cc = 0; cc < 8; ++cc) {
      int j = l + 32 * cc;
      unsigned flat = (((unsigned)(b * 256 + t)) << 8) | (unsigned)j;
      float x = Vs[(((size_t)b * 256) + t) * 256 + j] * 5000.f + gumbel_at(flat);
      vbuf[cc] = x;
      m = fmaxf(m, x);
    }
#pragma unroll
    for (int msk = 1; msk < 32; msk <<= 1) m = fmaxf(m, __shfl_xor(m, msk, 32));
    float se = 0.f, sj = 0.f;
#pragma unroll
    for (int cc = 0; cc < 8; ++cc) {
      int j = l + 32 * cc;
      float e = expf((vbuf[cc] - m) * 1e4f);  // /tau
      se += e;
      sj += e * (float)j;
    }
#pragma unroll
    for (int msk = 1; msk < 32; msk <<= 1) {
      se += __shfl_xor(se, msk, 32);
      sj += __shfl_xor(sj, msk, 32);
    }
    if (l == 0) {
      float fi = sj / se;
      idxOut[b * 256 + t] = fi;
      bool keep = (fi >= 0.f) && (fi <= 87.f) && (fi == truncf(fi));  // isin(arange(88))
      float mev = keep ? (fi + 1.f) : 0.f;
      mev = (mev != 0.f) ? 1.f : 0.f;  // me/me
      int ii = (int)fi;
      ii = ii < 0 ? 0 : (ii > 255 ? 255 : ii);
      me[b * 256 + t] = mev;
      ew[b * 256 + t] = vals[b * 256 + ii];
    }
  }
}

// ---------------------------------------------------------------------------
// Kernel 4: FC = X * me * ew ; emb = FC @ Wp^T + bp + posenc.  (K=116)
// ---------------------------------------------------------------------------
__global__ void k_embed(const float* __restrict__ X, const float* __restrict__ Wp,
                        const float* __restrict__ bp, const float* __restrict__ me,
                        const float* __restrict__ ew, float* __restrict__ emb) {
  extern __shared__ char smem[];
  float* sA = (float*)smem;       // 256 x 118 (scaled X rows)
  float* sW = sA + 256 * 118;     // Wp pair-interleaved [58][128][2]
  const int b = blockIdx.x, tid = threadIdx.x;
  for (int u = tid; u < 256 * 116; u += 256) {
    int t = u / 116, k = u % 116;
    float sc = me[b * 256 + t] * ew[b * 256 + t];
    sA[t * 118 + k] = X[(size_t)b * 29696 + u] * sc;
  }
  for (int u = tid; u < 128 * 116; u += 256) {
    int n = u / 116, k = u % 116;
    sW[(k >> 1) * 256 + (n << 1) + (k & 1)] = Wp[u];
  }
  __syncthreads();
  const int wave = tid >> 5, l = tid & 31, h = l >> 4, nl = l & 15;
  for (int tt = wave; tt < 16; tt += 8) {
    const int t0 = tt * 16;
    v8f acc[8] = {};
    for (int s = 0; s < 29; ++s) {
      const int k0 = 4 * s + 2 * h;
      v2f a = *(const v2f*)&sA[(t0 + nl) * 118 + k0];
      const float* wrow = &sW[(2 * s + h) * 256];
#pragma unroll
      for (int nt = 0; nt < 8; ++nt) {
        v2f bb = *(const v2f*)&wrow[(nt * 16 + nl) << 1];
        acc[nt] = wmma4(a, bb, acc[nt]);
      }
    }
#pragma unroll
    for (int nt = 0; nt < 8; ++nt)
#pragma unroll
      for (int v = 0; v < 8; ++v) {
        int n = nt * 16 + nl, t = t0 + v + 8 * h;
        float val = acc[nt][v] + bp[n];
        float ang = (float)t / powf(10000.f, (float)(n & ~1) / 128.f);
        val += (n & 1) ? cosf(ang) : sinf(ang);
        emb[(((size_t)b * 256) + t) * 128 + n] = val;
      }
  }
}

// ---------------------------------------------------------------------------
// Kernel 5: Q/K/V projections -> (B,H,T,DK). Q pre-scaled by 1/sqrt(DK). K=128
// ---------------------------------------------------------------------------
__global__ void k_qkv(const float* __restrict__ emb, const float* __restrict__ Wq,
                      const float* __restrict__ bq, const float* __restrict__ Wk,
                      const float* __restrict__ bk, const float* __restrict__ Wv,
                      const float* __restrict__ bv, float* __restrict__ Q,
                      float* __restrict__ Km, float* __restrict__ Vm) {
  extern __shared__ char smem[];
  float* sA = (float*)smem;       // 256 x 130
  float* sW = sA + 256 * 130;     // pair-interleaved [64][128][2]
  const int b = blockIdx.x, which = blockIdx.y, tid = threadIdx.x;
  const float* W = (which == 0) ? Wq : (which == 1) ? Wk : Wv;
  const float* bias = (which == 0) ? bq : (which == 1) ? bk : bv;
  float* Out = (which == 0) ? Q : (which == 1) ? Km : Vm;
  const float oscale = (which == 0) ? 0.25f : 1.f;  // 1/sqrt(16)

  for (int u = tid; u < 256 * 128; u += 256)
    sA[(u >> 7) * 130 + (u & 127)] = emb[(size_t)b * 32768 + u];
  for (int u = tid; u < 128 * 128; u += 256) {
    int n = u >> 7, k = u & 127;
    sW[(k >> 1) * 256 + (n << 1) + (k & 1)] = W[u];
  }
  __syncthreads();

  const int wave = tid >> 5, l = tid & 31, h = l >> 4, nl = l & 15;
  for (int tt = wave; tt < 16; tt += 8) {
    const int t0 = tt * 16;
    v8f acc[8] = {};
    for (int s = 0; s < 32; ++s) {
      const int k0 = 4 * s + 2 * h;
      v2f a = *(const v2f*)&sA[(t0 + nl) * 130 + k0];
      const float* wrow = &sW[(2 * s + h) * 256];
#pragma unroll
      for (int nt = 0; nt < 8; ++nt) {
        v2f bb = *(const v2f*)&wrow[(nt * 16 + nl) << 1];
        acc[nt] = wmma4(a, bb, acc[nt]);
      }
    }
#pragma unroll
    for (int nt = 0; nt < 8; ++nt)
#pragma unroll
      for (int v = 0; v < 8; ++v) {
        int n = nt * 16 + nl, t = t0 + v + 8 * h;
        float val = (acc[nt][v] + bias[n]) * oscale;
        int head = n >> 4, dk = n & 15;
        Out[(((size_t)(b * 8 + head) * 256) + t) * 16 + dk] = val;
      }
  }
}

// ---------------------------------------------------------------------------
// Kernel 6: fused attention per (b,h,16-query tile): QK^T -> softmax -> @V.
// One wave/block; scores K=16 = 4x4 exact; P re-enters A-layout via LDS.
// ---------------------------------------------------------------------------
__global__ void k_attn(const float* __restrict__ Q, const float* __restrict__ Km,
                       const float* __restrict__ Vm, float* __restrict__ ctx) {
  extern __shared__ char smem[];
  float* sP = (float*)smem;  // 16 x 258
  const int tt = blockIdx.x, bh = blockIdx.y;
  const int b = bh >> 3, head = bh & 7;
  const int l = threadIdx.x & 31, h = l >> 4, nl = l & 15;
  const int t0 = tt * 16;
  const float* Qb = Q  + (size_t)bh * 4096;
  const float* Kb = Km + (size_t)bh * 4096;
  const float* Vb = Vm + (size_t)bh * 4096;

  v2f af[4];
#pragma unroll
  for (int s = 0; s < 4; ++s)
    af[s] = *(const v2f*)&Qb[(t0 + nl) * 16 + 4 * s + 2 * h];

  v8f sc[16];
#pragma unroll
  for (int jt = 0; jt < 16; ++jt) {
    v8f z = {};
#pragma unroll
    for (int s = 0; s < 4; ++s) {
      v2f bb = *(const v2f*)&Kb[(jt * 16 + nl) * 16 + 4 * s + 2 * h];
      z = wmma4(af[s], bb, z);
    }
    sc[jt] = z;
  }
  // row softmax over 256 keys (rows live in 16-lane halves)
  float rmax[8], rsum[8];
#pragma unroll
  for (int v = 0; v < 8; ++v) {
    float m = sc[0][v];
#pragma unroll
    for (int jt = 1; jt < 16; ++jt) m = fmaxf(m, sc[jt][v]);
#pragma unroll
    for (int msk = 1; msk < 16; msk <<= 1) m = fmaxf(m, __shfl_xor(m, msk, 32));
    rmax[v] = m;
  }
#pragma unroll
  for (int v = 0; v < 8; ++v) {
    float s = 0.f;
#pragma unroll
    for (int jt = 0; jt < 16; ++jt) {
      float e = expf(sc[jt][v] - rmax[v]);
      sc[jt][v] = e;
      s += e;
    }
#pragma unroll
    for (int msk = 1; msk < 16; msk <<= 1) s += __shfl_xor(s, msk, 32);
    rsum[v] = s;
  }
#pragma unroll
  for (int jt = 0; jt < 16; ++jt)
#pragma unroll
    for (int v = 0; v < 8; ++v)
      sP[(v + 8 * h) * 258 + jt * 16 + nl] = sc[jt][v] / rsum[v];
  __syncthreads();

  v8f cacc = {};
  for (int s = 0; s < 64; ++s) {
    const int k0 = 4 * s + 2 * h;
    v2f a = *(const v2f*)&sP[nl * 258 + k0];
    v2f bb;
    bb.x = Vb[k0 * 16 + nl];
    bb.y = Vb[(k0 + 1) * 16 + nl];
    cacc = wmma4(a, bb, cacc);
  }
#pragma unroll
  for (int v = 0; v < 8; ++v) {
    int t = t0 + v + 8 * h;
    ctx[(((size_t)b * 256) + t) * 128 + head * 16 + nl] = cacc[v];
  }
}

// ---------------------------------------------------------------------------
// Kernel 7: out1 = LayerNorm(ctx @ Wo^T + bo + emb) * g1 + be1   (K=128)
// ---------------------------------------------------------------------------
__global__ void k_proj_ln1(const float* __restrict__ ctx, const float* __restrict__ Wo,
                           const float* __restrict__ bo, const float* __restrict__ emb,
                           const float* __restrict__ g1, const float* __restrict__ be1,
                           float* __restrict__ out1) {
  extern __shared__ char smem[];
  float* sA = (float*)smem;       // 256 x 130
  float* sW = sA + 256 * 130;     // [64][128][2]
  const int b = blockIdx.x, tid = threadIdx.x;
  for (int u = tid; u < 256 * 128; u += 256)
    sA[(u >> 7) * 130 + (u & 127)] = ctx[(size_t)b * 32768 + u];
  for (int u = tid; u < 128 * 128; u += 256) {
    int n = u >> 7, k = u & 127;
    sW[(k >> 1) * 256 + (n << 1) + (k & 1)] = Wo[u];
  }
  __syncthreads();

  const int wave = tid >> 5, l = tid & 31, h = l >> 4, nl = l & 15;
  for (int tt = wave; tt < 16; tt += 8) {
    const int t0 = tt * 16;
    v8f acc[8] = {};
    for (int s = 0; s < 32; ++s) {
      const int k0 = 4 * s + 2 * h;
      v2f a = *(const v2f*)&sA[(t0 + nl) * 130 + k0];
      const float* wrow = &sW[(2 * s + h) * 256];
#pragma unroll
      for (int nt = 0; nt < 8; ++nt) {
        v2f bb = *(const v2f*)&wrow[(nt * 16 + nl) << 1];
        acc[nt] = wmma4(a, bb, acc[nt]);
      }
    }
    float vals[8][8];
#pragma unroll
    for (int nt = 0; nt < 8; ++nt)
#pragma unroll
      for (int v = 0; v < 8; ++v) {
        int n = nt * 16 + nl, t = t0 + v + 8 * h;
        vals[nt][v] = acc[nt][v] + bo[n] + emb[(((size_t)b * 256) + t) * 128 + n];
      }
    float mu[8], rv[8];
#pragma unroll
    for (int v = 0; v < 8; ++v) {
      float a = 0.f, q = 0.f;
#pragma unroll
      for (int nt = 0; nt < 8; ++nt) {
        float x = vals[nt][v];
        a += x;
        q += x * x;
      }
#pragma unroll
      for (int m = 1; m < 16; m <<= 1) {
        a += __shfl_xor(a, m, 32);
        q += __shfl_xor(q, m, 32);
      }
      float mean = a * (1.f / 128.f);
      float var = q * (1.f / 128.f) - mean * mean;
      mu[v] = mean;
      rv[v] = rsqrtf(var + 1e-5f);
    }
#pragma unroll
    for (int nt = 0; nt < 8; ++nt)
#pragma unroll
      for (int v = 0; v < 8; ++v) {
        int n = nt * 16 + nl, t = t0 + v + 8 * h;
        float x = (vals[nt][v] - mu[v]) * rv[v];
        out1[(((size_t)b * 256) + t) * 128 + n] = x * g1[n] + be1[n];
      }
  }
}

// ---------------------------------------------------------------------------
// Kernel 8: hid = relu(out1 @ Wf1^T + bf1), N=512 in 4 chunks, W from L2.
// ---------------------------------------------------------------------------
__global__ void k_ff1(const float* __restrict__ out1, const float* __restrict__ Wf1,
                      const float* __restrict__ bf1, float* __restrict__ hid) {
  extern __shared__ char smem[];
  float* sA = (float*)smem;  // 256 x 130
  const int b = blockIdx.x, tid = threadIdx.x;
  for (int u = tid; u < 256 * 128; u += 256)
    sA[(u >> 7) * 130 + (u & 127)] = out1[(size_t)b * 32768 + u];
  __syncthreads();

  const int wave = tid >> 5, l = tid & 31, h = l >> 4, nl = l & 15;
  for (int tt = wave; tt < 16; tt += 8) {
    const int t0 = tt * 16;
    for (int nc = 0; nc < 4; ++nc) {
      v8f acc[8] = {};
      for (int s = 0; s < 32; ++s) {
        const int k0 = 4 * s + 2 * h;
        v2f a = *(const v2f*)&sA[(t0 + nl) * 130 + k0];
#pragma unroll
        for (int nt = 0; nt < 8; ++nt) {
          int n = nc * 128 + nt * 16 + nl;
          if (s == 0) __builtin_prefetch(&Wf1[n * 128], 0, 1);
          v2f bb = *(const v2f*)&Wf1[n * 128 + k0];
          acc[nt] = wmma4(a, bb, acc[nt]);
        }
      }
#pragma unroll
      for (int nt = 0; nt < 8; ++nt)
#pragma unroll
        for (int v = 0; v < 8; ++v) {
          int n = nc * 128 + nt * 16 + nl, t = t0 + v + 8 * h;
          hid[(((size_t)b * 256) + t) * 512 + n] = fmaxf(acc[nt][v] + bf1[n], 0.f);
        }
    }
  }
}

// ---------------------------------------------------------------------------
// Kernel 9: out = LayerNorm(hid @ Wf2^T + bf2 + out1) * g2 + be2   (K=512)
// A streamed from global (32KB/tile, L2-hot); Wf2 pair-interleaved in 256KB LDS.
// ---------------------------------------------------------------------------
__global__ void k_ff2_ln2(const float* __restrict__ hid, const float* __restrict__ Wf2,
                          const float* __restrict__ bf2, const float* __restrict__ out1,
                          const float* __restrict__ g2, const float* __restrict__ be2,
                          float* __restrict__ out) {
  extern __shared__ char smem[];
  float* sW = (float*)smem;  // [256][128][2] = 256KB
  const int b = blockIdx.x, tid = threadIdx.x;
  for (int u = tid; u < 128 * 512; u += 256) {
    int n = u >> 9, k = u & 511;
    sW[(k >> 1) * 256 + (n << 1) + (k & 1)] = Wf2[u];
  }
  __syncthreads();

  const int wave = tid >> 5, l = tid & 31, h = l >> 4, nl = l & 15;
  for (int tt = wave; tt < 16; tt += 8) {
    const int t0 = tt * 16;
    const float* Arow = &hid[(((size_t)b * 256) + t0 + nl) * 512];
    v8f acc[8] = {};
    for (int s = 0; s < 128; ++s) {
      const int k0 = 4 * s + 2 * h;
      v2f a = *(const v2f*)&Arow[k0];
      const float* wrow = &sW[(2 * s + h) * 256];
#pragma unroll
      for (int nt = 0; nt < 8; ++nt) {
        v2f bb = *(const v2f*)&wrow[(nt * 16 + nl) << 1];
        acc[nt] = wmma4(a, bb, acc[nt]);
      }
    }
    float vals[8][8];
#pragma unroll
    for (int nt = 0; nt < 8; ++nt)
#pragma unroll
      for (int v = 0; v < 8; ++v) {
        int n = nt * 16 + nl, t = t0 + v + 8 * h;
        vals[nt][v] = acc[nt][v] + bf2[n] + out1[(((size_t)b * 256) + t) * 128 + n];
      }
    float mu[8], rv[8];
#pragma unroll
    for (int v = 0; v < 8; ++v) {
      float a = 0.f, q = 0.f;
#pragma unroll
      for (int nt = 0; nt < 8; ++nt) {
        float x = vals[nt][v];
        a += x;
        q += x * x;
      }
#pragma unroll
      for (int m = 1; m < 16; m <<= 1) {
        a += __shfl_xor(a, m, 32);
        q += __shfl_xor(q, m, 32);
      }
      float mean = a * (1.f / 128.f);
      float var = q * (1.f / 128.f) - mean * mean;
      mu[v] = mean;
      rv[v] = rsqrtf(var + 1e-5f);
    }
#pragma unroll
    for (int nt = 0; nt < 8; ++nt)
#pragma unroll
      for (int v = 0; v < 8; ++v) {
        int n = nt * 16 + nl, t = t0 + v + 8 * h;
        float x = (vals[nt][v] - mu[v]) * rv[v];
        out[(((size_t)b * 256) + t) * 128 + n] = x * g2[n] + be2[n];
      }
  }
}

// ---------------------------------------------------------------------------
extern "C" void kernel_launch(void* const* d_in, const int* in_sizes, int n_in,
                              void* d_out, int out_size, void* d_ws, size_t ws_size,
                              hipStream_t stream) {
  (void)in_sizes; (void)n_in; (void)out_size; (void)ws_size;
  const float* X    = (const float*)d_in[0];
  const float* W1   = (const float*)d_in[1];
  const float* b1   = (const float*)d_in[2];
  const float* g_bn = (const float*)d_in[3];
  const float* bebn = (const float*)d_in[4];
  const float* W2   = (const float*)d_in[5];
  const float* b2   = (const float*)d_in[6];
  const float* Wp   = (const float*)d_in[7];
  const float* bp   = (const float*)d_in[8];
  const float* Wq   = (const float*)d_in[9];
  const float* bq   = (const float*)d_in[10];
  const float* Wk   = (const float*)d_in[11];
  const float* bk   = (const float*)d_in[12];
  const float* Wv   = (const float*)d_in[13];
  const float* bv   = (const float*)d_in[14];
  const float* Wo   = (const float*)d_in[15];
  const float* bo   = (const float*)d_in[16];
  const float* g1   = (const float*)d_in[17];
  const float* be1  = (const float*)d_in[18];
  const float* Wf1  = (const float*)d_in[19];
  const float* bf1  = (const float*)d_in[20];
  const float* Wf2  = (const float*)d_in[21];
  const float* bf2  = (const float*)d_in[22];
  const float* g2   = (const float*)d_in[23];
  const float* be2  = (const float*)d_in[24];

  float* outp = (float*)d_out;            // (B,T,D)   524288
  float* gm   = outp + 524288;            // (B,T,T)  1048576
  float* idxo = gm + 1048576;             // (B,T)       4096

  float* ws     = (float*)d_ws;
  float* wsV    = ws;                     // 1048576
  float* wsVs   = wsV + 1048576;          // 1048576
  float* wsVal  = wsVs + 1048576;         // 4096
  float* wsMe   = wsVal + 4096;           // 4096
  float* wsEw   = wsMe + 4096;            // 4096
  float* wsEmb  = wsEw + 4096;            // 524288
  float* wsQ    = wsEmb + 524288;         // 524288
  float* wsK    = wsQ + 524288;           // 524288
  float* wsVv   = wsK + 524288;           // 524288
  float* wsCtx  = wsVv + 524288;          // 524288
  float* wsO1   = wsCtx + 524288;         // 524288
  float* wsHid  = wsO1 + 524288;          // 2097152

  const size_t lds_gm   = (120 + 256 * 118 + 58 * 256) * 4;        // 180704
  const size_t lds_eig  = 65536 * 4 + 4 * 128 * 4 + 256 * 4 * 2;   // 266240
  const size_t lds_k116 = (256 * 118 + 58 * 256) * 4;              // 180224
  const size_t lds_k128 = (256 * 130 + 64 * 256) * 4;              // 198656
  const size_t lds_attn = 16 * 258 * 4;                            //  16512
  const size_t lds_ff1  = 256 * 130 * 4;                           // 133120
  const size_t lds_ff2  = 256 * 256 * 4;                           // 262144

  k_gmpool<<<dim3(256, 16), 256, lds_gm, stream>>>(X, W1, b1, g_bn, bebn, W2, b2, gm);
  k_eigh<<<16, 256, lds_eig, stream>>>(gm, wsV, wsVs, wsVal);
  k_group<<<16, 256, 0, stream>>>(wsVs, wsVal, wsMe, wsEw, idxo);
  k_embed<<<16, 256, lds_k116, stream>>>(X, Wp, bp, wsMe, wsEw, wsEmb);
  k_qkv<<<dim3(16, 3), 256, lds_k128, stream>>>(wsEmb, Wq, bq, Wk, bk, Wv, bv,
                                                wsQ, wsK, wsVv);
  k_attn<<<dim3(16, 128), 32, lds_attn, stream>>>(wsQ, wsK, wsVv, wsCtx);
  k_proj_ln1<<<16, 256, lds_k128, stream>>>(wsCtx, Wo, bo, wsEmb, g1, be1, wsO1);
  k_ff1<<<16, 256, lds_ff1, stream>>>(wsO1, Wf1, bf1, wsHid);
  k_ff2_ln2<<<16, 256, lds_ff2, stream>>>(wsHid, Wf2, bf2, wsO1, g2, be2, outp);
}